// BipartiteSAGE_85615878078794
// MI455X (gfx1250) — compile-verified
//
#include <hip/hip_runtime.h>

// ---------------------------------------------------------------------------
// Problem constants (match reference)
// ---------------------------------------------------------------------------
#define N_OCC   20000
#define N_SKILL 50000
#define N_EDGE  600000
#define D_IN    256
#define HID     128
#define OUTF    64

typedef __attribute__((ext_vector_type(2))) float v2f;
typedef __attribute__((ext_vector_type(8))) float v8f;

// ---------------------------------------------------------------------------
// WMMA GEMM:  out[M,N] = act( scale?(A1 @ W1) (+ A2 @ W2) + bias )
//
//   * V_WMMA_F32_16X16X4_F32 (f32-native, matches f32 reference precision)
//   * each wave computes a 16(M) x 64(N) strip with FOUR accumulators:
//       - one A fragment feeds 4 WMMAs  -> 1.25 loads per WMMA
//       - 4 independent D->C accumulate chains hide WMMA latency
//   * block = 128 threads = 4 waves = 4 row tiles -> 64(M) x 64(N) per block
//   * K, N are template constants; A and B use ONE base pointer each with
//     pure immediate offsets (max ~49 KB < 24-bit IOFFSET) -> zero address
//     arithmetic in the inner loop
//   * W pre-transposed (WT[n*K+k]) so each B fragment is one global_load_b64
//   * SAGE mean folded in: scale accumulators by 1/max(cnt,1) between the
//     Wl product and the fused Wr product (row scaling commutes with @W)
//
// Fragment layouts (CDNA5 ISA 7.12.2, wave32):
//   A (16x4 f32):  lane L holds A[row0 + (L&15)][k0 + 2*(L>>4) + i], i=0,1
//   B (4x16 f32):  lane L holds W[k0 + 2*(L>>4) + i][col + (L&15)]
//   C/D (16x16):   c[r] = D[row0 + r + 8*(L>>4)][col + (L&15)]
// N multiples of 64; tail row-tiles handled by uniform whole-wave early-out,
// so EXEC is all ones for every executed WMMA (hardware requirement).
// ---------------------------------------------------------------------------
template <int K, int N, bool FUSE>
__global__ void __launch_bounds__(128)
gemm_sage(const float* __restrict__ A1, const float* __restrict__ WT1,
          const float* __restrict__ A2, const float* __restrict__ WT2,
          const unsigned* __restrict__ cnt, const float* __restrict__ bias,
          float* __restrict__ out, int M, int do_relu)
{
    const int lane = threadIdx.x & 31;
    const int wave = threadIdx.x >> 5;
    const int hi   = lane >> 4;        // half-wave select
    const int lm   = lane & 15;        // A row / B,C,D column within tile
    const int rt   = blockIdx.x * 4 + wave;   // row tile handled by this wave
    if (rt * 16 >= M) return;                 // uniform per-wave tail guard
    const int row0 = rt * 16;
    const int col0 = blockIdx.y * 64;         // wave spans 4 col tiles

    v8f c0 = {0.f,0.f,0.f,0.f,0.f,0.f,0.f,0.f};
    v8f c1 = c0, c2 = c0, c3 = c0;

    {
        const float* ap = A1  + (size_t)(row0 + lm) * K + 2 * hi;
        const float* bp = WT1 + (size_t)(col0 + lm) * K + 2 * hi;
        #pragma unroll 8
        for (int k = 0; k < K / 4; ++k) {
            v2f a  = *(const v2f*)(ap + 4 * k);
            v2f b0 = *(const v2f*)(bp + 4 * k);
            v2f b1 = *(const v2f*)(bp + 4 * k + 16 * K);
            v2f b2 = *(const v2f*)(bp + 4 * k + 32 * K);
            v2f b3 = *(const v2f*)(bp + 4 * k + 48 * K);
            c0 = __builtin_amdgcn_wmma_f32_16x16x4_f32(false, a, false, b0, (short)0, c0, false, false);
            c1 = __builtin_amdgcn_wmma_f32_16x16x4_f32(false, a, false, b1, (short)0, c1, false, false);
            c2 = __builtin_amdgcn_wmma_f32_16x16x4_f32(false, a, false, b2, (short)0, c2, false, false);
            c3 = __builtin_amdgcn_wmma_f32_16x16x4_f32(false, a, false, b3, (short)0, c3, false, false);
        }
    }

    if (FUSE) {
        // mean-aggregate scaling of the first product (SAGE lin_l path)
        #pragma unroll
        for (int r = 0; r < 8; ++r) {
            const float inv = 1.0f / fmaxf((float)cnt[row0 + r + 8 * hi], 1.0f);
            c0[r] *= inv; c1[r] *= inv; c2[r] *= inv; c3[r] *= inv;
        }
        // fused second product: + A2 @ W2 (SAGE lin_r path)
        const float* ap = A2  + (size_t)(row0 + lm) * K + 2 * hi;
        const float* bp = WT2 + (size_t)(col0 + lm) * K + 2 * hi;
        #pragma unroll 8
        for (int k = 0; k < K / 4; ++k) {
            v2f a  = *(const v2f*)(ap + 4 * k);
            v2f b0 = *(const v2f*)(bp + 4 * k);
            v2f b1 = *(const v2f*)(bp + 4 * k + 16 * K);
            v2f b2 = *(const v2f*)(bp + 4 * k + 32 * K);
            v2f b3 = *(const v2f*)(bp + 4 * k + 48 * K);
            c0 = __builtin_amdgcn_wmma_f32_16x16x4_f32(false, a, false, b0, (short)0, c0, false, false);
            c1 = __builtin_amdgcn_wmma_f32_16x16x4_f32(false, a, false, b1, (short)0, c1, false, false);
            c2 = __builtin_amdgcn_wmma_f32_16x16x4_f32(false, a, false, b2, (short)0, c2, false, false);
            c3 = __builtin_amdgcn_wmma_f32_16x16x4_f32(false, a, false, b3, (short)0, c3, false, false);
        }
    }

    // epilogue: bias (+ReLU) and store all 4 tiles
    float* orow = out + (size_t)(row0 + 8 * hi) * N + col0 + lm;
    const float bv0 = bias[col0 + lm];
    const float bv1 = bias[col0 + 16 + lm];
    const float bv2 = bias[col0 + 32 + lm];
    const float bv3 = bias[col0 + 48 + lm];
    #pragma unroll
    for (int r = 0; r < 8; ++r) {
        float v0 = c0[r] + bv0, v1 = c1[r] + bv1, v2 = c2[r] + bv2, v3 = c3[r] + bv3;
        if (do_relu) {
            v0 = fmaxf(v0, 0.f); v1 = fmaxf(v1, 0.f);
            v2 = fmaxf(v2, 0.f); v3 = fmaxf(v3, 0.f);
        }
        float* p = orow + (size_t)r * N;
        p[0] = v0; p[16] = v1; p[32] = v2; p[48] = v3;
    }
}

// ---------------------------------------------------------------------------
// Small helpers
// ---------------------------------------------------------------------------

// WT[n*K + k] = W[k*N + n]  (tiny matrices; coalesced read)
__global__ void __launch_bounds__(256)
transpose_w(const float* __restrict__ W, float* __restrict__ WT, int K, int N)
{
    int i = blockIdx.x * blockDim.x + threadIdx.x;
    if (i < K * N) {
        int k = i / N, n = i - k * N;
        WT[(size_t)n * K + k] = W[i];
    }
}

__device__ inline void atomic_add_f32(float* p, float v)
{
    (void)__hip_atomic_fetch_add(p, v, __ATOMIC_RELAXED, __HIP_MEMORY_SCOPE_AGENT);
}

// degree counts for both node types (edges reused by both conv layers)
__global__ void __launch_bounds__(256)
count_deg(const int* __restrict__ e_occ, const int* __restrict__ e_skill,
          unsigned* __restrict__ cnt_occ, unsigned* __restrict__ cnt_skill, int E)
{
    int e = blockIdx.x * blockDim.x + threadIdx.x;
    if (e < E) {
        atomicAdd(&cnt_occ[e_occ[e]], 1u);
        atomicAdd(&cnt_skill[e_skill[e]], 1u);
    }
}

// agg[dst[e], :] += hsrc[src[e], :]   (HID=128 features, float4 per thread)
__global__ void __launch_bounds__(256)
scatter_add(const float* __restrict__ hsrc, const int* __restrict__ src,
            const int* __restrict__ dst, float* __restrict__ agg, int E)
{
    const long long tid = (long long)blockIdx.x * blockDim.x + threadIdx.x;
    const int e = (int)(tid >> 5);          // 32 float4-chunks per edge
    const int f = ((int)tid & 31) << 2;     // feature offset
    if (e < E) {
        const int s = src[e];
        const int d = dst[e];
        const float4 v = *(const float4*)(hsrc + (size_t)s * HID + f);
        float* p = agg + (size_t)d * HID + f;
        atomic_add_f32(p + 0, v.x);
        atomic_add_f32(p + 1, v.y);
        atomic_add_f32(p + 2, v.z);
        atomic_add_f32(p + 3, v.w);
    }
}

// ---------------------------------------------------------------------------
// Orchestration
// ---------------------------------------------------------------------------
extern "C" void kernel_launch(void* const* d_in, const int* in_sizes, int n_in,
                              void* d_out, int out_size, void* d_ws, size_t ws_size,
                              hipStream_t stream)
{
    (void)in_sizes; (void)n_in; (void)out_size; (void)ws_size;

    const float* x_occ    = (const float*)d_in[0];
    const float* x_skill  = (const float*)d_in[1];
    const int*   e_occ    = (const int*)  d_in[2];
    const int*   e_skill  = (const int*)  d_in[3];
    const float* pW_occ   = (const float*)d_in[4];
    const float* pb_occ   = (const float*)d_in[5];
    const float* pW_skill = (const float*)d_in[6];
    const float* pb_skill = (const float*)d_in[7];
    const float* c1_os_Wl = (const float*)d_in[8];
    const float* c1_os_bl = (const float*)d_in[9];
    const float* c1_os_Wr = (const float*)d_in[10];
    const float* c1_so_Wl = (const float*)d_in[11];
    const float* c1_so_bl = (const float*)d_in[12];
    const float* c1_so_Wr = (const float*)d_in[13];
    const float* c2_os_Wl = (const float*)d_in[14];
    const float* c2_os_bl = (const float*)d_in[15];
    const float* c2_os_Wr = (const float*)d_in[16];
    const float* c2_so_Wl = (const float*)d_in[17];
    const float* c2_so_bl = (const float*)d_in[18];
    const float* c2_so_Wr = (const float*)d_in[19];

    // ---- workspace layout (floats) ----
    float* ws = (float*)d_ws;
    float* h_occ     = ws;                                    // 20000*128
    float* h_skill   = h_occ     + (size_t)N_OCC   * HID;     // 50000*128
    float* o1        = h_skill   + (size_t)N_SKILL * HID;     // 20000*128
    float* s1        = o1        + (size_t)N_OCC   * HID;     // 50000*128
    float* agg_occ   = s1        + (size_t)N_SKILL * HID;     // 20000*128
    float* agg_skill = agg_occ   + (size_t)N_OCC   * HID;     // 50000*128
    unsigned* cnt_occ   = (unsigned*)(agg_skill + (size_t)N_SKILL * HID); // 20000
    unsigned* cnt_skill = cnt_occ + N_OCC;                                 // 50000
    float* wt = (float*)(cnt_skill + N_SKILL);
    float* pWt_occ   = wt;                 wt += D_IN * HID;
    float* pWt_skill = wt;                 wt += D_IN * HID;
    float* t1_os_Wl  = wt;                 wt += HID * HID;
    float* t1_os_Wr  = wt;                 wt += HID * HID;
    float* t1_so_Wl  = wt;                 wt += HID * HID;
    float* t1_so_Wr  = wt;                 wt += HID * HID;
    float* t2_os_Wl  = wt;                 wt += HID * OUTF;
    float* t2_os_Wr  = wt;                 wt += HID * OUTF;
    float* t2_so_Wl  = wt;                 wt += HID * OUTF;
    float* t2_so_Wr  = wt;                 wt += HID * OUTF;

    float* o2 = (float*)d_out;                 // [20000,64]
    float* s2 = o2 + (size_t)N_OCC * OUTF;     // [50000,64]

    const dim3 blk128(128), blk256(256);
    const int rt_occ   = (N_OCC / 16 + 3) / 4;    // row-tile groups of 4 waves
    const int rt_skill = (N_SKILL / 16 + 3) / 4;

    // ---- zero accumulation scratch (deterministic per call) ----
    hipMemsetAsync(cnt_occ, 0, sizeof(unsigned) * (N_OCC + N_SKILL), stream);
    hipMemsetAsync(agg_occ, 0, sizeof(float) * (size_t)(N_OCC + N_SKILL) * HID, stream);

    // ---- transpose all GEMM weights (tiny) ----
    transpose_w<<<(D_IN * HID + 255) / 256, blk256, 0, stream>>>(pW_occ,   pWt_occ,   D_IN, HID);
    transpose_w<<<(D_IN * HID + 255) / 256, blk256, 0, stream>>>(pW_skill, pWt_skill, D_IN, HID);
    transpose_w<<<(HID * HID + 255) / 256, blk256, 0, stream>>>(c1_os_Wl, t1_os_Wl, HID, HID);
    transpose_w<<<(HID * HID + 255) / 256, blk256, 0, stream>>>(c1_os_Wr, t1_os_Wr, HID, HID);
    transpose_w<<<(HID * HID + 255) / 256, blk256, 0, stream>>>(c1_so_Wl, t1_so_Wl, HID, HID);
    transpose_w<<<(HID * HID + 255) / 256, blk256, 0, stream>>>(c1_so_Wr, t1_so_Wr, HID, HID);
    transpose_w<<<(HID * OUTF + 255) / 256, blk256, 0, stream>>>(c2_os_Wl, t2_os_Wl, HID, OUTF);
    transpose_w<<<(HID * OUTF + 255) / 256, blk256, 0, stream>>>(c2_os_Wr, t2_os_Wr, HID, OUTF);
    transpose_w<<<(HID * OUTF + 255) / 256, blk256, 0, stream>>>(c2_so_Wl, t2_so_Wl, HID, OUTF);
    transpose_w<<<(HID * OUTF + 255) / 256, blk256, 0, stream>>>(c2_so_Wr, t2_so_Wr, HID, OUTF);

    // ---- degree counts (shared by conv1 & conv2) ----
    count_deg<<<(N_EDGE + 255) / 256, blk256, 0, stream>>>(
        e_occ, e_skill, cnt_occ, cnt_skill, N_EDGE);

    // ---- input projections + ReLU ----
    gemm_sage<D_IN, HID, false><<<dim3(rt_occ, HID / 64), blk128, 0, stream>>>(
        x_occ, pWt_occ, nullptr, nullptr, nullptr, pb_occ, h_occ, N_OCC, 1);
    gemm_sage<D_IN, HID, false><<<dim3(rt_skill, HID / 64), blk128, 0, stream>>>(
        x_skill, pWt_skill, nullptr, nullptr, nullptr, pb_skill, h_skill, N_SKILL, 1);

    // ---- conv1 aggregation ----
    const int sg_blocks = (int)(((long long)N_EDGE * 32 + 255) / 256);
    scatter_add<<<sg_blocks, blk256, 0, stream>>>(h_occ,   e_occ,   e_skill, agg_skill, N_EDGE);
    scatter_add<<<sg_blocks, blk256, 0, stream>>>(h_skill, e_skill, e_occ,   agg_occ,   N_EDGE);

    // ---- conv1 linear: s1 = relu(mean@Wl + bl + h_skill@Wr), same for o1 ----
    gemm_sage<HID, HID, true><<<dim3(rt_skill, HID / 64), blk128, 0, stream>>>(
        agg_skill, t1_os_Wl, h_skill, t1_os_Wr, cnt_skill, c1_os_bl, s1, N_SKILL, 1);
    gemm_sage<HID, HID, true><<<dim3(rt_occ, HID / 64), blk128, 0, stream>>>(
        agg_occ, t1_so_Wl, h_occ, t1_so_Wr, cnt_occ, c1_so_bl, o1, N_OCC, 1);

    // ---- conv2 aggregation (reuse agg buffers) ----
    hipMemsetAsync(agg_occ, 0, sizeof(float) * (size_t)(N_OCC + N_SKILL) * HID, stream);
    scatter_add<<<sg_blocks, blk256, 0, stream>>>(o1, e_occ,   e_skill, agg_skill, N_EDGE);
    scatter_add<<<sg_blocks, blk256, 0, stream>>>(s1, e_skill, e_occ,   agg_occ,   N_EDGE);

    // ---- conv2 linear (no ReLU) straight into d_out ----
    gemm_sage<HID, OUTF, true><<<dim3(rt_skill, OUTF / 64), blk128, 0, stream>>>(
        agg_skill, t2_os_Wl, s1, t2_os_Wr, cnt_skill, c2_os_bl, s2, N_SKILL, 0);
    gemm_sage<HID, OUTF, true><<<dim3(rt_occ, OUTF / 64), blk128, 0, stream>>>(
        agg_occ, t2_so_Wl, o1, t2_so_Wr, cnt_occ, c2_so_bl, o2, N_OCC, 0);
}